// DeconvDft2dLayer_65335042507377
// MI455X (gfx1250) — compile-verified
//
#include <hip/hip_runtime.h>
#include <hip/hip_bf16.h>

// DeconvDft2d via matmul-DFT on CDNA5 WMMA (v_wmma_f32_16x16x4_f32),
// with double-buffered Tensor Data Mover panel staging (tensor_load_to_lds
// overlapped with WMMA compute, synchronized via s_wait_tensorcnt), and D#
// padding producing bank-conflict-free LDS tiles (pitch 33 / 65).
//
// Pipeline (all f32):
//   stage1: per-image r2c row DFT   : P0 = X(512x512) * W1(512x448)       [2 GEMMs]
//   stage2: complex col DFT         : P1 = F2(768x512) * P0               [4 GEMMs]
//   filter: P1 *= G (closed-form reciprocal-spectrum product)             [pointwise]
//   stage3: complex inverse col DFT : P0 = F3(768x768) * P1               [4 GEMMs]
//   stage4: c2r row inverse DFT     : out = P0re*Vc + P0im*Vs             [2 GEMMs]

typedef float v2f __attribute__((ext_vector_type(2)));
typedef float v8f __attribute__((ext_vector_type(8)));
typedef unsigned int u32x4 __attribute__((ext_vector_type(4)));
typedef int i32x4 __attribute__((ext_vector_type(4)));
typedef int i32x8 __attribute__((ext_vector_type(8)));

#define BATCH 64
#define HX 512        // input rows/cols
#define HP 768        // padded size
#define NF 385        // rfft bins (768/2+1)
#define NP 448        // padded bins (multiple of 64)
#define TWOPI 6.28318530717958647692f

// ---------------------------------------------------------------------------
// TDM 2D tile load: global(row-major, pitch elements) -> LDS with row padding.
// D# layout per CDNA5 ISA ch.8: group0 {count, lds_addr, global_addr, type=2},
// group1 {data_size=4B, pad_enable, pad_interval/amount, dims, tile, stride}.
// pad_interval_code: 4 => pad after every 32 DWORDs, 5 => every 64 DWORDs;
// pad amount code 0 => 1 DWORD, giving LDS pitches 33 / 65.
// ---------------------------------------------------------------------------
__device__ __forceinline__ void tdm_load_2d(unsigned lds_byte, const void* gptr,
                                            unsigned pitch_elems, unsigned rows,
                                            unsigned tile_w, unsigned tile_h,
                                            unsigned pad_interval_code)
{
  unsigned long long ga = (unsigned long long)(size_t)gptr;
  u32x4 g0;
  g0[0] = 1u;                                              // count=1 (valid user D#)
  g0[1] = lds_byte;                                        // lds_addr (bytes)
  g0[2] = (unsigned)(ga & 0xFFFFFFFFull);                  // global_addr[31:0]
  g0[3] = (unsigned)((ga >> 32) & 0x01FFFFFFull) | (2u << 30); // addr[56:32] | type=2

  i32x8 g1;
  g1[0] = (int)((2u << 16) | (1u << 20) | (pad_interval_code << 22)); // 4B, pad_en, amount=1dw
  g1[1] = (int)((pitch_elems & 0xFFFFu) << 16);            // tensor_dim0[15:0]
  g1[2] = (int)(((pitch_elems >> 16) & 0xFFFFu) | ((rows & 0xFFFFu) << 16));
  g1[3] = (int)(((rows >> 16) & 0xFFFFu) | ((tile_w & 0xFFFFu) << 16)); // tile_dim0
  g1[4] = (int)(tile_h & 0xFFFFu);                         // tile_dim1 (tile_dim2=0)
  g1[5] = (int)pitch_elems;                                // tensor_dim0_stride[31:0]
  g1[6] = 0;
  g1[7] = 0;

  i32x4 z4 = {0, 0, 0, 0};
#if defined(__clang_major__) && (__clang_major__ >= 23)
  i32x8 z8 = {0, 0, 0, 0, 0, 0, 0, 0};
  __builtin_amdgcn_tensor_load_to_lds(g0, g1, z4, z4, z8, 0);
#else
  __builtin_amdgcn_tensor_load_to_lds(g0, g1, z4, z4, 0);
#endif
}

// ---------------------------------------------------------------------------
// Generic batched GEMM:  C = alpha * A(MxK) * B(KxN) [+ C if beta]
// Row-major, lda=K, ldb=N, ldc=N. M,N multiples of 64; K multiple of 32.
// Block: 128 threads = 4 waves; each block does a 64x64 C tile;
// each wave a 16x64 strip via 4 accumulators of v_wmma_f32_16x16x4_f32.
// TDM double-buffering: wave 0 issues next panels while all waves compute;
// one workgroup barrier per K step.
// ---------------------------------------------------------------------------
__global__ __launch_bounds__(128)
void wmma_gemm(float* __restrict__ C, const float* __restrict__ A,
               const float* __restrict__ B,
               int M, int N, int K, long sA, long sB, long sC,
               float alpha, int beta)
{
  __shared__ float As[2][64][33];   // 64 rows x 32 K, pad 1 -> pitch 33
  __shared__ float Bs[2][32][65];   // 32 K rows x 64 cols, pad 1 -> pitch 65

  const int bz = blockIdx.z;
  A += (size_t)sA * bz;
  B += (size_t)sB * bz;
  C += (size_t)sC * bz;

  const int rowBlk = blockIdx.y * 64;
  const int colBlk = blockIdx.x * 64;
  const int tid  = threadIdx.x;
  const int wave = tid >> 5;
  const int lane = tid & 31;
  const int m    = lane & 15;        // A: row within 16; B/C/D: col within 16
  const int kh   = (lane >> 4) << 1; // 0 or 2 : K sub-offset per lane half

  // LDS byte addresses for the TDM descriptors (flat LDS addr, low 32 bits)
  unsigned asAddr[2], bsAddr[2];
  asAddr[0] = (unsigned)(size_t)(&As[0][0][0]);
  asAddr[1] = (unsigned)(size_t)(&As[1][0][0]);
  bsAddr[0] = (unsigned)(size_t)(&Bs[0][0][0]);
  bsAddr[1] = (unsigned)(size_t)(&Bs[1][0][0]);

  v8f acc0 = {}, acc1 = {}, acc2 = {}, acc3 = {};

  const int nsteps = K >> 5;

  // prologue: stage first panels into buffer 0
  if (wave == 0) {
    tdm_load_2d(asAddr[0], A + (size_t)rowBlk * K, (unsigned)K, 64u, 32u, 64u, 4u);
    tdm_load_2d(bsAddr[0], B + colBlk, (unsigned)N, 32u, 64u, 32u, 5u);
  }

  int cur = 0;
  for (int s = 0; s < nsteps; ++s) {
    if (wave == 0) __builtin_amdgcn_s_wait_tensorcnt(0);
    __syncthreads();   // publish buf[cur]; proves buf[cur^1] fully consumed

    const int nxt = cur ^ 1;
    if (wave == 0 && s + 1 < nsteps) {
      const int kb = (s + 1) << 5;
      // A panel: 64 rows x 32 dwords, row pitch K; pad every 32 dwords
      tdm_load_2d(asAddr[nxt], A + (size_t)rowBlk * K + kb, (unsigned)K, 64u, 32u, 64u, 4u);
      // B panel: 32 rows x 64 dwords, row pitch N; pad every 64 dwords
      tdm_load_2d(bsAddr[nxt], B + (size_t)kb * N + colBlk, (unsigned)N, 32u, 64u, 32u, 5u);
    } else if (wave == 1 && s + 2 < nsteps) {
      __builtin_prefetch(A + (size_t)rowBlk * K + ((s + 2) << 5), 0, 1);
    } else if (wave == 2 && s + 2 < nsteps) {
      __builtin_prefetch(B + (size_t)((s + 2) << 5) * N + colBlk, 0, 1);
    }

    const float (*Ac)[33] = As[cur];
    const float (*Bc)[65] = Bs[cur];
    #pragma unroll
    for (int kk = 0; kk < 32; kk += 4) {
      v2f a;
      a.x = Ac[wave * 16 + m][kk + kh];
      a.y = Ac[wave * 16 + m][kk + kh + 1];
      v2f b0, b1, b2, b3;
      b0.x = Bc[kk + kh][m];      b0.y = Bc[kk + kh + 1][m];
      b1.x = Bc[kk + kh][16 + m]; b1.y = Bc[kk + kh + 1][16 + m];
      b2.x = Bc[kk + kh][32 + m]; b2.y = Bc[kk + kh + 1][32 + m];
      b3.x = Bc[kk + kh][48 + m]; b3.y = Bc[kk + kh + 1][48 + m];
      acc0 = __builtin_amdgcn_wmma_f32_16x16x4_f32(false, a, false, b0, (short)0, acc0, false, false);
      acc1 = __builtin_amdgcn_wmma_f32_16x16x4_f32(false, a, false, b1, (short)0, acc1, false, false);
      acc2 = __builtin_amdgcn_wmma_f32_16x16x4_f32(false, a, false, b2, (short)0, acc2, false, false);
      acc3 = __builtin_amdgcn_wmma_f32_16x16x4_f32(false, a, false, b3, (short)0, acc3, false, false);
    }
    cur = nxt;
  }

  // store: lane<16 -> rows 0..7 in acc[0..7]; lane>=16 -> rows 8..15
  const int rowh = (lane >> 4) * 8;
  #pragma unroll
  for (int r = 0; r < 8; ++r) {
    int gm = rowBlk + wave * 16 + rowh + r;
    size_t base = (size_t)gm * N + colBlk + m;
    float v0 = alpha * acc0[r];
    float v1 = alpha * acc1[r];
    float v2 = alpha * acc2[r];
    float v3 = alpha * acc3[r];
    if (beta) {
      v0 += C[base];
      v1 += C[base + 16];
      v2 += C[base + 32];
      v3 += C[base + 48];
    }
    C[base]      = v0;
    C[base + 16] = v1;
    C[base + 32] = v2;
    C[base + 48] = v3;
  }
}

// ---------------------------------------------------------------------------
// Init kernels: DFT matrices via exact integer phase mod 768
// ---------------------------------------------------------------------------
__global__ void init_w1(float* W1c, float* W1s)  // [512 x 448] r2c row DFT, x offset +128
{
  int k = blockIdx.x * 64 + threadIdx.x; // 0..447
  int x = blockIdx.y;                    // 0..511
  size_t o = (size_t)x * NP + k;
  if (k >= NF) { W1c[o] = 0.f; W1s[o] = 0.f; return; }
  int ph = (int)(((long)k * (x + 128)) % HP);
  float th = TWOPI * (float)ph / (float)HP;
  W1c[o] = cosf(th);
  W1s[o] = -sinf(th);
}

__global__ void init_f2(float* F2re, float* F2im) // [768 x 512] fwd col DFT, y offset +128
{
  int yp = blockIdx.x * 64 + threadIdx.x; // 0..511
  int ky = blockIdx.y;                    // 0..767
  size_t o = (size_t)ky * HX + yp;
  int ph = (int)(((long)ky * (yp + 128)) % HP);
  float th = TWOPI * (float)ph / (float)HP;
  F2re[o] = cosf(th);
  F2im[o] = -sinf(th);
}

__global__ void init_f3(float* F3re, float* F3im) // [768 x 768] inv col DFT, scaled 1/768
{
  int ky = blockIdx.x * 64 + threadIdx.x; // 0..767
  int y  = blockIdx.y;                    // 0..767
  size_t o = (size_t)y * HP + ky;
  int ph = (int)(((long)ky * y) % HP);
  float th = TWOPI * (float)ph / (float)HP;
  const float s = 1.0f / (float)HP;
  F3re[o] = s * cosf(th);
  F3im[o] = s * sinf(th);
}

__global__ void init_v(float* Vc, float* Vs) // [448 x 768] c2r row inverse, weights folded
{
  int x = blockIdx.x * 64 + threadIdx.x; // 0..767
  int k = blockIdx.y;                    // 0..447
  size_t o = (size_t)k * HP + x;
  if (k >= NF) { Vc[o] = 0.f; Vs[o] = 0.f; return; }
  float wk = (k == 0 || k == HP / 2) ? (1.0f / (float)HP) : (2.0f / (float)HP);
  int ph = (int)(((long)k * x) % HP);
  float th = TWOPI * (float)ph / (float)HP;
  Vc[o] = wk * cosf(th);
  Vs[o] = -wk * sinf(th);
}

// closed-form rfft2 of the 3x3 filter padded to 768x768 (9-term sum)
__device__ inline float2 hval(const float* wl, int a, int b)
{
  float re = 0.f, im = 0.f;
  #pragma unroll
  for (int i = 0; i < 3; ++i)
    #pragma unroll
    for (int j = 0; j < 3; ++j) {
      int ph = (a * i + b * j) % HP;
      float th = TWOPI * (float)ph / (float)HP;
      float c = wl[i * 3 + j];
      re += c * cosf(th);
      im -= c * sinf(th);
    }
  return make_float2(re, im);
}
__device__ inline float2 cinv(float2 h)
{
  float d = h.x * h.x + h.y * h.y;
  return make_float2(h.x / d, -h.y / d);
}
__device__ inline float2 cmul2(float2 p, float2 q)
{
  return make_float2(p.x * q.x - p.y * q.y, p.x * q.y + p.y * q.x);
}

__global__ void init_g(float* Gre, float* Gim, const float* __restrict__ w)
{
  int kx = blockIdx.x * 64 + threadIdx.x; // 0..447
  int ky = blockIdx.y;                    // 0..767
  float wl[9];
  #pragma unroll
  for (int t = 0; t < 9; ++t) wl[t] = w[t];
  int a2 = (HP - ky) % HP;                       // gm2/gm4 row reflection
  int b3 = ((NF - kx) % NF + NF) % NF;           // gm3/gm4 rfft-axis reflection
  float2 g = cmul2(cmul2(cinv(hval(wl, ky, kx)), cinv(hval(wl, a2, kx))),
                   cmul2(cinv(hval(wl, ky, b3)), cinv(hval(wl, a2, b3))));
  size_t o = (size_t)ky * NP + kx;
  Gre[o] = g.x;
  Gim[o] = g.y;
}

// pointwise complex filter multiply on P1 (bins >= NF are zero, stay zero)
__global__ void cmul_filter(float* Pre, float* Pim,
                            const float* __restrict__ Gre,
                            const float* __restrict__ Gim)
{
  int j = blockIdx.x * 256 + threadIdx.x;              // 0 .. 768*448-1
  size_t off = (size_t)blockIdx.y * (HP * NP) + j;
  float a = Pre[off], b = Pim[off];
  float c = Gre[j],  d = Gim[j];
  Pre[off] = a * c - b * d;
  Pim[off] = a * d + b * c;
}

// ---------------------------------------------------------------------------
static void launch_gemm(float* C, const float* A, const float* B,
                        int M, int N, int K, long sA, long sB, long sC,
                        float alpha, int beta, hipStream_t s)
{
  dim3 g(N / 64, M / 64, BATCH);
  wmma_gemm<<<g, 128, 0, s>>>(C, A, B, M, N, K, sA, sB, sC, alpha, beta);
}

extern "C" void kernel_launch(void* const* d_in, const int* in_sizes, int n_in,
                              void* d_out, int out_size, void* d_ws, size_t ws_size,
                              hipStream_t stream)
{
  const float* x = (const float*)d_in[0]; // [64,512,512]
  const float* w = (const float*)d_in[1]; // [3,3]
  float* out = (float*)d_out;             // [64,768,768]
  float* ws = (float*)d_ws;

  const size_t plane = (size_t)BATCH * HP * NP;
  float* P0re = ws;                       float* P0im = P0re + plane;
  float* P1re = P0im + plane;             float* P1im = P1re + plane;
  float* Gre  = P1im + plane;             float* Gim  = Gre + (size_t)HP * NP;
  float* W1c  = Gim + (size_t)HP * NP;    float* W1s  = W1c + (size_t)HX * NP;
  float* F2re = W1s + (size_t)HX * NP;    float* F2im = F2re + (size_t)HP * HX;
  float* F3re = F2im + (size_t)HP * HX;   float* F3im = F3re + (size_t)HP * HP;
  float* Vc   = F3im + (size_t)HP * HP;   float* Vs   = Vc + (size_t)NP * HP;

  // constants (recomputed each call; deterministic, no cached state)
  init_w1<<<dim3(NP / 64, HX), 64, 0, stream>>>(W1c, W1s);
  init_f2<<<dim3(HX / 64, HP), 64, 0, stream>>>(F2re, F2im);
  init_f3<<<dim3(HP / 64, HP), 64, 0, stream>>>(F3re, F3im);
  init_v <<<dim3(HP / 64, NP), 64, 0, stream>>>(Vc, Vs);
  init_g <<<dim3(NP / 64, HP), 64, 0, stream>>>(Gre, Gim, w);

  const long sIn  = (long)HX * HX;  // 512*512 per image
  const long sP   = (long)HP * NP;  // complex-plane pitch per image
  const long sOut = (long)HP * HP;  // 768*768 per image

  // stage 1: r2c row DFT (writes rows 0..511 of each 768-row plane)
  launch_gemm(P0re, x, W1c, HX, NP, HX, sIn, 0, sP, 1.f, 0, stream);
  launch_gemm(P0im, x, W1s, HX, NP, HX, sIn, 0, sP, 1.f, 0, stream);

  // stage 2: forward column DFT (complex GEMM as 4 real GEMMs, K=512 valid rows)
  launch_gemm(P1re, F2re, P0re, HP, NP, HX, 0, sP, sP,  1.f, 0, stream);
  launch_gemm(P1re, F2im, P0im, HP, NP, HX, 0, sP, sP, -1.f, 1, stream);
  launch_gemm(P1im, F2re, P0im, HP, NP, HX, 0, sP, sP,  1.f, 0, stream);
  launch_gemm(P1im, F2im, P0re, HP, NP, HX, 0, sP, sP,  1.f, 1, stream);

  // frequency-domain filter
  cmul_filter<<<dim3((HP * NP) / 256, BATCH), 256, 0, stream>>>(P1re, P1im, Gre, Gim);

  // stage 3: inverse column DFT (scale 1/768 folded into F3)
  launch_gemm(P0re, F3re, P1re, HP, NP, HP, 0, sP, sP,  1.f, 0, stream);
  launch_gemm(P0re, F3im, P1im, HP, NP, HP, 0, sP, sP, -1.f, 1, stream);
  launch_gemm(P0im, F3re, P1im, HP, NP, HP, 0, sP, sP,  1.f, 0, stream);
  launch_gemm(P0im, F3im, P1re, HP, NP, HP, 0, sP, sP,  1.f, 1, stream);

  // stage 4: c2r row inverse DFT (hermitian weights folded into Vc/Vs)
  launch_gemm(out, P0re, Vc, HP, HP, NP, sP, 0, sOut, 1.f, 0, stream);
  launch_gemm(out, P0im, Vs, HP, HP, NP, sP, 0, sOut, 1.f, 1, stream);
}